// KernelEncodingLayer_17935783428711
// MI455X (gfx1250) — compile-verified
//
#include <hip/hip_runtime.h>
#include <math.h>

#define NKEYS   8192
#define NBINS   128
#define NFREQ   64
#define NKER    3
#define HEADD   128
#define MTERMS  8
#define K2      (NFREQ * 2 * MTERMS)     // 1024
#define CHUNK   32                       // k2 values staged per LDS chunk
#define FPC     (CHUNK / (2 * MTERMS))   // 2 freqs per chunk
#define NCHUNK  (K2 / CHUNK)             // 32
#define ROWS    128                      // key rows per block
#define FSTRIDE 36                       // sF row stride (dwords): 36 mod 64 = 36 -> conflict-free b64 A loads
#define PSTRIDE 288                      // sW k-pair stride (dwords): 288 mod 64 = 32 -> conflict-free b64 B loads

typedef float v2f __attribute__((ext_vector_type(2)));
typedef float v8f __attribute__((ext_vector_type(8)));

// ---------------------------------------------------------------------------
// Kernel 1: fold kernels/kappa/mu/weight into Fourier coefficients
//   Wc[(f*16 + 2m + 0)*NBINS + b] = sum_k w * e^{-kappa} * I_m(kappa) * (m?2:1) * cos(m*mu_eff)
//   Wc[(f*16 + 2m + 1)*NBINS + b] = ...                                        * sin(m*mu_eff)
// I_m(x) via power series (kappa<=1 -> 6 terms ample; I_8(1)/I_0(1) ~ 8e-8).
// ---------------------------------------------------------------------------
__global__ __launch_bounds__(256) void vm_coeff_kernel(
    const float* __restrict__ ref_angles,
    const float* __restrict__ mu,
    const float* __restrict__ kappa,
    const float* __restrict__ weight,
    float* __restrict__ Wc)
{
    int idx = blockIdx.x * blockDim.x + threadIdx.x;
    if (idx >= NBINS * NFREQ) return;
    int b = idx / NFREQ;
    int f = idx - b * NFREQ;
    float ra = ref_angles[f];

    float CA[MTERMS], CB[MTERMS];
#pragma unroll
    for (int m = 0; m < MTERMS; ++m) { CA[m] = 0.0f; CB[m] = 0.0f; }

    for (int kk = 0; kk < NKER; ++kk) {
        int o = (b * NFREQ + f) * NKER + kk;
        float mu_eff = mu[o] + ra;
        float kap = kappa[o];
        float w   = weight[o];
        float emk = expf(-kap);
        float half = 0.5f * kap;
        float h2 = half * half;
        float pw = 1.0f;                 // (kap/2)^m / m!
        float cmu = cosf(mu_eff), smu = sinf(mu_eff);
        float cr = 1.0f, ci = 0.0f;      // cos(m*mu_eff), sin(m*mu_eff)
#pragma unroll
        for (int m = 0; m < MTERMS; ++m) {
            float t = pw, s = pw;        // I_m(kap) power series
#pragma unroll
            for (int j = 1; j <= 6; ++j) {
                t *= h2 / (float)(j * (m + j));
                s += t;
            }
            float coef = w * emk * s * (m == 0 ? 1.0f : 2.0f);
            CA[m] += coef * cr;
            CB[m] += coef * ci;
            pw *= half / (float)(m + 1);
            float nr = cr * cmu - ci * smu;
            ci = cr * smu + ci * cmu;
            cr = nr;
        }
    }
#pragma unroll
    for (int m = 0; m < MTERMS; ++m) {
        Wc[(f * 16 + 2 * m + 0) * NBINS + b] = CA[m];
        Wc[(f * 16 + 2 * m + 1) * NBINS + b] = CB[m];
    }
}

// ---------------------------------------------------------------------------
// Kernel 2: fused feature-expansion + f32 WMMA GEMM
//   logits[n,b] = sum_{k2} X[n,k2] * Wc[k2,b] + bias[b]
// Block: 256 threads (8 waves), 128 key-rows x 128 bins per block.
// W chunk staged K-pair-interleaved so each B fragment is one ds_load_b64
// landing directly in an even-aligned VGPR pair (no packing moves).
// ---------------------------------------------------------------------------
__global__ __launch_bounds__(256) void vm_gemm_kernel(
    const float* __restrict__ Kmat,
    const float* __restrict__ Wc,
    const float* __restrict__ bias,
    float* __restrict__ out)
{
    __shared__ float sF[ROWS * FSTRIDE];            // features [row][k2-in-chunk]
    __shared__ float sW[(CHUNK / 2) * PSTRIDE];     // coeffs   [k-pair][2*bin + parity]

    const int tid  = threadIdx.x;
    const int wave = tid >> 5;        // 0..7
    const int lane = tid & 31;
    const int h    = lane >> 4;       // half-wave select (K pair / row+8)
    const int l    = lane & 15;       // M / N index within tile
    const int row0 = blockIdx.x * ROWS;
    const int m0   = wave * 16;       // this wave's row offset in block

    v8f acc[8];
#pragma unroll
    for (int t = 0; t < 8; ++t) acc[t] = (v8f)0.0f;

    for (int ch = 0; ch < NCHUNK; ++ch) {
        __syncthreads();   // protect previous chunk's LDS reads

        // --- stage W chunk (32 x 128 floats), pair-interleaved ---
        const int k2base = ch * CHUNK;
#pragma unroll
        for (int i = tid; i < CHUNK * NBINS; i += 256) {
            int kr = i >> 7;          // 0..31
            int bb = i & 127;
            sW[(kr >> 1) * PSTRIDE + 2 * bb + (kr & 1)] =
                Wc[(k2base + kr) * NBINS + bb];
        }

        // --- compute feature chunk: one (n, f) pair per thread ---
        {
            int n  = tid >> 1;            // 0..127
            int fl = tid & 1;             // 0..FPC-1
            int f  = ch * FPC + fl;
            v2f xy = *reinterpret_cast<const v2f*>(
                &Kmat[(size_t)(row0 + n) * HEADD + 2 * f]);
            float x = xy.x, y = xy.y;
            float mag = sqrtf(x * x + y * y);
            float inv = (mag > 1e-30f) ? (1.0f / mag) : 0.0f;
            float c = (mag > 1e-30f) ? (x * inv) : 1.0f;   // cos(angle)
            float s = y * inv;                             // sin(angle)
            float re = mag, im = 0.0f;                     // mag * e^{i m a}
            float* dst = &sF[n * FSTRIDE + fl * (2 * MTERMS)];
#pragma unroll
            for (int m = 0; m < MTERMS; ++m) {
                dst[2 * m]     = re;
                dst[2 * m + 1] = im;
                float nre = re * c - im * s;
                im = re * s + im * c;
                re = nre;
            }
        }
        __syncthreads();

        // --- WMMA over this chunk: K advances 4 per v_wmma_f32_16x16x4_f32 ---
#pragma unroll
        for (int k0 = 0; k0 < CHUNK; k0 += 4) {
            v2f av = *reinterpret_cast<const v2f*>(
                &sF[(m0 + l) * FSTRIDE + k0 + 2 * h]);     // A[m][k0+2h..+1]
            const float* wp = &sW[((k0 >> 1) + h) * PSTRIDE + 2 * l];
#pragma unroll
            for (int t = 0; t < 8; ++t) {
                v2f bv = *reinterpret_cast<const v2f*>(&wp[t * 32]);
                acc[t] = __builtin_amdgcn_wmma_f32_16x16x4_f32(
                    false, av, false, bv, (short)0, acc[t], false, false);
            }
        }
    }

    // --- epilogue: bias + store (C/D layout: VGPR r -> rows r / r+8) ---
#pragma unroll
    for (int t = 0; t < 8; ++t) {
        float bv = bias[t * 16 + l];
#pragma unroll
        for (int r = 0; r < 8; ++r) {
            int row = row0 + m0 + r + h * 8;
            out[(size_t)row * NBINS + t * 16 + l] = acc[t][r] + bv;
        }
    }
}

extern "C" void kernel_launch(void* const* d_in, const int* in_sizes, int n_in,
                              void* d_out, int out_size, void* d_ws, size_t ws_size,
                              hipStream_t stream) {
    const float* K      = (const float*)d_in[0];
    const float* ref    = (const float*)d_in[1];
    const float* mu     = (const float*)d_in[2];
    const float* kappa  = (const float*)d_in[3];
    const float* weight = (const float*)d_in[4];
    const float* bias   = (const float*)d_in[5];
    float* out = (float*)d_out;
    float* Wc  = (float*)d_ws;   // K2 x NBINS f32 = 512 KB

    vm_coeff_kernel<<<(NBINS * NFREQ + 255) / 256, 256, 0, stream>>>(ref, mu, kappa, weight, Wc);
    vm_gemm_kernel<<<NKEYS / ROWS, 256, 0, stream>>>(K, Wc, bias, out);
}